// GaussianVectorQuantizer_28174985461968
// MI455X (gfx1250) — compile-verified
//
#include <hip/hip_runtime.h>
#include <math.h>

#define B_    8
#define NPTS  1024
#define NDIM  64
#define K_    16
#define S_    512

#define TN    32          // n-rows per workgroup
#define LSTR  513         // padded LDS stride for logits tile (bank-conflict-free)
#define ZSTR  65          // padded LDS stride for ze tile

#define PREC_OFF (B_*NPTS*NDIM)               // 524288
#define PROB_OFF (PREC_OFF + 1)               // 524289
#define LOGP_OFF (PROB_OFF + B_*K_*NPTS*S_)   // 67633153

typedef float v2f __attribute__((ext_vector_type(2)));
typedef float v8f __attribute__((ext_vector_type(8)));

__device__ __forceinline__ float waveMax(float v) {
#pragma unroll
  for (int o = 16; o > 0; o >>= 1) v = fmaxf(v, __shfl_xor(v, o, 32));
  return v;
}
__device__ __forceinline__ float waveSum(float v) {
#pragma unroll
  for (int o = 16; o > 0; o >>= 1) v += __shfl_xor(v, o, 32);
  return v;
}

// ---------------------------------------------------------------------------
// Init: zero zq (atomic accumulation target), write precision_q, precompute
// row norms ||ze_n||^2 (ws[0..8192)) and ||book_s||^2 (ws[8192..16384)).
// ---------------------------------------------------------------------------
__global__ __launch_bounds__(256) void gvq_init(const float* __restrict__ ze,
                                                const float* __restrict__ books,
                                                const float* __restrict__ lpq,
                                                float* __restrict__ out,
                                                float* __restrict__ ws) {
  const int bid = blockIdx.x, t = threadIdx.x;
  if (bid < 2048) {
    out[bid * 256 + t] = 0.0f;                    // zq region: 2048*256 = 524288
    if (bid == 0 && t == 0)
      out[PREC_OFF] = 0.5f / fmaxf(__expf(lpq[0]), 1e-10f);
  } else {
    const int r = (bid - 2048) * 256 + t;         // 0..16383
    const float* src = (r < 8192) ? (ze + r * NDIM) : (books + (r - 8192) * NDIM);
    float s = 0.0f;
#pragma unroll 8
    for (int i = 0; i < NDIM; ++i) { float x = src[i]; s += x * x; }
    ws[r] = s;
  }
}

// ---------------------------------------------------------------------------
// Fused main kernel: one workgroup = (b, k, 32 n-rows).
//  Phase 1: logits[32][512] via V_WMMA_F32_16X16X4_F32 -> LDS
//  Phase 2: per-row (wave-private) softmax -> prob/log_prob (NT stores),
//           gumbel-softmax numerators -> LDS (in place), 1/sum -> rowScale
//  Phase 3: zq tile = enc[32x512] x books[512x64] via WMMA, scaled by
//           c_probs[b,k]*rowScale, accumulated with global f32 atomics.
// ---------------------------------------------------------------------------
__global__ __launch_bounds__(256) void gvq_main(const float* __restrict__ ze,
                                                const float* __restrict__ c_probs,
                                                const float* __restrict__ books,
                                                const float* __restrict__ lpq,
                                                const float* __restrict__ gum,
                                                float* __restrict__ out,
                                                const float* __restrict__ ws) {
  extern __shared__ float smem[];
  float* Lg       = smem;                       // [TN][LSTR]
  float* zeA      = Lg + TN * LSTR;             // [TN][ZSTR]
  float* ze2row   = zeA + TN * ZSTR;            // [TN]
  float* rowScale = ze2row + TN;                // [TN]

  const int t    = threadIdx.x;
  const int lane = t & 31;
  const int w    = t >> 5;
  const int lo   = lane & 15;
  const int hi   = lane >> 4;
  const int n0   = blockIdx.x * TN;
  const int k    = blockIdx.y;
  const int b    = blockIdx.z;

  const float pq = 0.5f / fmaxf(__expf(lpq[0]), 1e-10f);
  const float* bk = books + k * S_ * NDIM;

  // Stage ze tile (coalesced) + row norms.
  {
    const float* src = ze + (b * NPTS + n0) * NDIM;
#pragma unroll
    for (int i = 0; i < 8; ++i) {
      int idx = t + 256 * i;
      zeA[(idx >> 6) * ZSTR + (idx & 63)] = src[idx];
    }
    if (t < TN) ze2row[t] = ws[b * NPTS + n0 + t];
  }
  __syncthreads();

  // ---- Phase 1: distance GEMM (full f32 precision) ----
  // A-fragments for both 16-row groups, all 16 K-steps, kept in registers.
  v2f aF[2][16];
#pragma unroll
  for (int tr = 0; tr < 2; ++tr)
#pragma unroll
    for (int kk = 0; kk < 16; ++kk) {
      int row = tr * 16 + lo;
      int kd  = 4 * kk + 2 * hi;
      aF[tr][kk] = (v2f){zeA[row * ZSTR + kd], zeA[row * ZSTR + kd + 1]};
    }

#pragma unroll 1
  for (int st = 0; st < 4; ++st) {              // wave owns s in [w*64, w*64+64)
    const int s0   = w * 64 + st * 16;
    const int srow = s0 + lo;
    v8f acc0 = {}; v8f acc1 = {};
#pragma unroll
    for (int kk = 0; kk < 16; ++kk) {
      int kd = 4 * kk + 2 * hi;
      const float* bp = bk + srow * NDIM + kd;
      v2f bF = (v2f){bp[0], bp[1]};
      acc0 = __builtin_amdgcn_wmma_f32_16x16x4_f32(false, aF[0][kk], false, bF,
                                                   (short)0, acc0, false, false);
      acc1 = __builtin_amdgcn_wmma_f32_16x16x4_f32(false, aF[1][kk], false, bF,
                                                   (short)0, acc1, false, false);
    }
    const float bk2v = ws[8192 + k * S_ + srow];   // ||book_s||^2, s = s0+lo = D col
#pragma unroll
    for (int j = 0; j < 8; ++j) {
      int r0 = j + 8 * hi;
      Lg[r0 * LSTR + s0 + lo]        = (2.0f * acc0[j] - ze2row[r0]      - bk2v) * pq;
      Lg[(16 + r0) * LSTR + s0 + lo] = (2.0f * acc1[j] - ze2row[16 + r0] - bk2v) * pq;
    }
  }
  __syncthreads();

  // ---- Phase 2: softmax + gumbel-softmax, one row per wave (no barriers) ----
  const int rowBase = (b * K_ + k) * NPTS + n0;
#pragma unroll 1
  for (int c = 0; c < 4; ++c) {
    const int row = c * 8 + w;
    float* Lr = &Lg[row * LSTR];
    const int gbase = (rowBase + row) * S_;

    float m = -3.4e38f;
#pragma unroll
    for (int i = 0; i < 16; ++i) m = fmaxf(m, Lr[lane + 32 * i]);
    m = waveMax(m);

    float s = 0.0f;
#pragma unroll
    for (int i = 0; i < 16; ++i) s += __expf(Lr[lane + 32 * i] - m);
    s = waveSum(s);
    const float ls = __logf(s);

#pragma unroll
    for (int i = 0; i < 16; ++i) {
      int col = lane + 32 * i;
      float lp = Lr[col] - m - ls;
      __builtin_nontemporal_store(__expf(lp), &out[PROB_OFF + gbase + col]);
      __builtin_nontemporal_store(lp,         &out[LOGP_OFF + gbase + col]);
    }

    float m2 = -3.4e38f;
#pragma unroll
    for (int i = 0; i < 16; ++i) {
      int col = lane + 32 * i;
      float u = __builtin_nontemporal_load(&gum[gbase + col]);
      float g = -__logf(-__logf(u + 1e-10f) + 1e-10f);
      float y = (Lr[col] + g) * 2.0f;            // /TEMPERATURE (0.5)
      Lr[col] = y;
      m2 = fmaxf(m2, y);
    }
    m2 = waveMax(m2);

    float s2 = 0.0f;
#pragma unroll
    for (int i = 0; i < 16; ++i) {
      int col = lane + 32 * i;
      float e = __expf(Lr[col] - m2);
      Lr[col] = e;                               // unnormalized encodings
      s2 += e;
    }
    s2 = waveSum(s2);
    if (lane == 0) rowScale[row] = 1.0f / s2;
  }
  __syncthreads();

  // ---- Phase 3: zq GEMM, enc[32x512] x books[512x64], K-depth 512 ----
  const int tRow = w >> 2, tCol = w & 3;         // 2x4 grid of 16x16 tiles
  v8f zacc = {};
#pragma unroll 4
  for (int kk = 0; kk < 128; ++kk) {
    int sK = 4 * kk + 2 * hi;
    const float* Er = &Lg[(tRow * 16 + lo) * LSTR + sK];
    v2f a = (v2f){Er[0], Er[1]};
    const float* bp = bk + sK * NDIM + tCol * 16 + lo;
    v2f bb = (v2f){bp[0], bp[NDIM]};
    zacc = __builtin_amdgcn_wmma_f32_16x16x4_f32(false, a, false, bb,
                                                 (short)0, zacc, false, false);
  }
  const float cp = c_probs[b * K_ + k];
#pragma unroll
  for (int j = 0; j < 8; ++j) {
    int row = tRow * 16 + j + 8 * hi;
    int d   = tCol * 16 + lo;
    float val = zacc[j] * cp * rowScale[row];
    atomicAdd(&out[(b * NPTS + n0 + row) * NDIM + d], val);
  }
}

// ---------------------------------------------------------------------------
extern "C" void kernel_launch(void* const* d_in, const int* in_sizes, int n_in,
                              void* d_out, int out_size, void* d_ws, size_t ws_size,
                              hipStream_t stream) {
  const float* ze    = (const float*)d_in[0];
  const float* cprob = (const float*)d_in[1];
  const float* books = (const float*)d_in[2];
  const float* lpq   = (const float*)d_in[3];
  const float* gum   = (const float*)d_in[4];
  float* out = (float*)d_out;
  float* ws  = (float*)d_ws;    // needs 64 KB: ze2[8192] + book2[8192]

  gvq_init<<<2112, 256, 0, stream>>>(ze, books, lpq, out, ws);

  dim3 grid(NPTS / TN, K_, B_);                  // (32, 16, 8) = 4096 WGs
  size_t lds = (size_t)(TN * LSTR + TN * ZSTR + 2 * TN) * sizeof(float); // ~74 KB
  gvq_main<<<grid, 256, lds, stream>>>(ze, cprob, books, lpq, gum, out, ws);
}